// DKVMNMODEL_71305047048801
// MI455X (gfx1250) — compile-verified
//
#include <hip/hip_runtime.h>
#include <math.h>

// DKVMN: B=256, S=256, DQ=64, DV=128, M=50, F=128
#define BB     256
#define SS     256
#define NROWS  (BB * SS)     // 65536
#define DQDIM  64
#define DVDIM  128
#define MMEM   50
#define MPAD   64
#define FDIM   128

typedef float v2f __attribute__((ext_vector_type(2)));
typedef float v8f __attribute__((ext_vector_type(8)));

// V_WMMA_F32_16X16X4_F32: D(16x16 f32) = A(16x4 f32) * B(4x16 f32) + C
__device__ __forceinline__ v8f wmma_k4(v2f a, v2f b, v8f c) {
  return __builtin_amdgcn_wmma_f32_16x16x4_f32(false, a, false, b, (short)0, c,
                                               false, false);
}

__device__ __forceinline__ float sigmoidf_(float x) {
  return 1.0f / (1.0f + expf(-x));
}

// ---- async global->LDS staging support (guarded; falls back to sync copies)
#if defined(__has_builtin)
# if __has_builtin(__builtin_amdgcn_global_load_async_to_lds_b128)
#  define ASYNC_MODE 2
# elif __has_builtin(__builtin_amdgcn_global_load_async_to_lds_b32)
#  define ASYNC_MODE 1
# endif
#endif
#ifndef ASYNC_MODE
# define ASYNC_MODE 0
#endif

#if ASYNC_MODE
// From hipcc's diagnostic: param0 of the b128 builtin is
// 'int __attribute__((vector_size(16))) __device__ *'  (AS1, non-const, int4).
typedef int v4i_ __attribute__((vector_size(16)));
typedef __attribute__((address_space(1))) v4i_* g128_t;
typedef __attribute__((address_space(3))) v4i_* l128_t;
typedef __attribute__((address_space(1))) int*  g32_t;
typedef __attribute__((address_space(3))) int*  l32_t;

__device__ __forceinline__ g128_t as_g128(const void* p) {
  return (g128_t)(unsigned long long)p;
}
__device__ __forceinline__ l128_t as_l128(void* p) {
  return (l128_t)(unsigned int)(unsigned long long)p;
}
__device__ __forceinline__ g32_t as_g32(const void* p) {
  return (g32_t)(unsigned long long)p;
}
__device__ __forceinline__ l32_t as_l32(void* p) {
  return (l32_t)(unsigned int)(unsigned long long)p;
}
#endif

__device__ __forceinline__ void wait_async0() {
#if ASYNC_MODE
# if __has_builtin(__builtin_amdgcn_s_wait_asynccnt)
  __builtin_amdgcn_s_wait_asynccnt(0);
# else
  asm volatile("s_wait_asynccnt 0x0" ::: "memory");
# endif
#endif
}

// Stage one timestep's operands (w:64 slots incl pad, e:128, a:128) into LDS.
// Async when available: issuing wave's ASYNCcnt is drained at next loop-top
// wait_async0(); the workgroup barrier then publishes LDS to all waves.
__device__ __forceinline__ void stage_step(
    int tid, size_t row, float* wsh, float* esh, float* ash,
    const float* __restrict__ wbuf, const float* __restrict__ erase,
    const float* __restrict__ addb) {
#if ASYNC_MODE == 2
  if (tid < 16) {
    __builtin_amdgcn_global_load_async_to_lds_b128(
        as_g128(wbuf + row * MPAD + tid * 4), as_l128(wsh + tid * 4), 0, 0);
  } else if (tid < 64) {
    // idle lanes
  } else if (tid < 96) {
    __builtin_amdgcn_global_load_async_to_lds_b128(
        as_g128(erase + row * DVDIM + (tid - 64) * 4),
        as_l128(esh + (tid - 64) * 4), 0, 0);
  } else if (tid < 128) {
    __builtin_amdgcn_global_load_async_to_lds_b128(
        as_g128(addb + row * DVDIM + (tid - 96) * 4),
        as_l128(ash + (tid - 96) * 4), 0, 0);
  }
#elif ASYNC_MODE == 1
  for (int i = tid; i < 320; i += 256) {
    if (i < 64)
      __builtin_amdgcn_global_load_async_to_lds_b32(
          as_g32(wbuf + row * MPAD + i), as_l32(wsh + i), 0, 0);
    else if (i < 192)
      __builtin_amdgcn_global_load_async_to_lds_b32(
          as_g32(erase + row * DVDIM + (i - 64)), as_l32(esh + (i - 64)), 0, 0);
    else
      __builtin_amdgcn_global_load_async_to_lds_b32(
          as_g32(addb + row * DVDIM + (i - 192)), as_l32(ash + (i - 192)), 0, 0);
  }
#else
  for (int i = tid; i < 320; i += 256) {
    if (i < 64)       wsh[i]       = wbuf[row * MPAD + i];
    else if (i < 192) esh[i - 64]  = erase[row * DVDIM + (i - 64)];
    else              ash[i - 192] = addb[row * DVDIM + (i - 192)];
  }
#endif
}

// ---------------------------------------------------------------- K0: zero
__global__ void k0_zero(float* acc) {
  if (threadIdx.x < 2) acc[threadIdx.x] = 0.0f;
}

// ----------------------------------------------- K1: w = softmax(q @ K^T)
__global__ __launch_bounds__(256) void k1_softmax_w(
    const int* __restrict__ qd, const float* __restrict__ qtab,
    const float* __restrict__ keym, float* __restrict__ wbuf) {
  __shared__ float skey[MMEM][DQDIM];  // 12.8 KB
  for (int i = threadIdx.x; i < MMEM * DQDIM; i += 256)
    skey[i / DQDIM][i % DQDIM] = keym[i];
  __syncthreads();

  int row = blockIdx.x * 256 + threadIdx.x;
  if (row >= NROWS) return;
  const float* q = qtab + (size_t)qd[row] * DQDIM;

  float acc[MMEM];
#pragma unroll
  for (int m = 0; m < MMEM; ++m) acc[m] = 0.0f;
  for (int k = 0; k < DQDIM; ++k) {
    float qk = q[k];
#pragma unroll
    for (int m = 0; m < MMEM; ++m) acc[m] = fmaf(qk, skey[m][k], acc[m]);
  }
  float mx = acc[0];
#pragma unroll
  for (int m = 1; m < MMEM; ++m) mx = fmaxf(mx, acc[m]);
  float s = 0.0f;
#pragma unroll
  for (int m = 0; m < MMEM; ++m) { acc[m] = expf(acc[m] - mx); s += acc[m]; }
  float inv = 1.0f / s;
  float* wr = wbuf + (size_t)row * MPAD;
#pragma unroll
  for (int m = 0; m < MMEM; ++m) wr[m] = acc[m] * inv;
}

// ------------------- K2: erase = sigmoid(qa@W_e + b_e), add = tanh(qa@W_a + b_a)
#define XPAD (DVDIM + 4)  // padded LDS stride -> lanes 0..15 hit distinct banks
__global__ __launch_bounds__(256) void k2_erase_add(
    const int* __restrict__ qad, const float* __restrict__ qatab,
    const float* __restrict__ We, const float* __restrict__ be,
    const float* __restrict__ Wa, const float* __restrict__ ba,
    float* __restrict__ erase, float* __restrict__ addb) {
  __shared__ float X[16][XPAD];
  const int row0 = blockIdx.x * 16;
  const int tid  = threadIdx.x;

  {  // gather 16 rows of qa_emb into LDS: 16 threads/row, 8 floats each
    int rl = tid >> 4, c0 = (tid & 15) * 8;
    const float* src = qatab + (size_t)qad[row0 + rl] * DVDIM + c0;
#pragma unroll
    for (int j = 0; j < 8; ++j) X[rl][c0 + j] = src[j];
  }
  __syncthreads();

  const int wave  = tid >> 5, lane = tid & 31;
  const int n0    = wave * 16;
  const int mrow  = lane & 15;          // A row / B col within tile
  const int khalf = (lane >> 4) * 2;    // lanes 16-31 carry K+2,K+3

  for (int which = 0; which < 2; ++which) {
    const float* W    = which ? Wa : We;
    const float* bias = which ? ba : be;
    float*       out  = which ? addb : erase;
    v8f c = {};
    for (int ks = 0; ks < DVDIM; ks += 4) {
      v2f a, b;
      a.x = X[mrow][ks + khalf];
      a.y = X[mrow][ks + khalf + 1];
      b.x = W[(size_t)(ks + khalf) * DVDIM + n0 + mrow];
      b.y = W[(size_t)(ks + khalf + 1) * DVDIM + n0 + mrow];
      c = wmma_k4(a, b, c);
    }
    const int mbase = (lane >> 4) * 8;  // C layout: VGPR r -> M = r (+8 hi lanes)
    const int n     = n0 + mrow;
#pragma unroll
    for (int r = 0; r < 8; ++r) {
      float v = c[r] + bias[n];
      v = which ? tanhf(v) : sigmoidf_(v);
      out[(size_t)(row0 + mbase + r) * DVDIM + n] = v;
    }
  }
}

// ------------- K3: sequential scan, one block per batch b, async double-buffer
__global__ __launch_bounds__(256) void k3_scan(
    const int* __restrict__ qd, const float* __restrict__ initv,
    const float* __restrict__ wbuf, const float* __restrict__ erase,
    const float* __restrict__ addb, float* __restrict__ reads) {
  __shared__ float mem[MMEM][DVDIM];  // 25.6 KB resident state
  __shared__ float wsh[2][MPAD];
  __shared__ float esh[2][DVDIM];
  __shared__ float ash[2][DVDIM];
  const int b = blockIdx.x, tid = threadIdx.x;

  for (int i = tid; i < MMEM * DVDIM; i += 256)
    mem[i / DVDIM][i % DVDIM] = initv[i];

  // prologue: stage t=0 into buffer 0
  stage_step(tid, (size_t)b * SS, wsh[0], esh[0], ash[0], wbuf, erase, addb);

  for (int t = 0; t < SS; ++t) {
    const size_t row = (size_t)b * SS + t;
    const int cur = t & 1;
    const float flag = (qd[row] >= 1) ? 1.0f : 0.0f;

    wait_async0();     // drain this wave's async copies (nop for non-issuers)
    __syncthreads();   // publish buffer `cur`; also fences prev update + mem init

    // stage t+1 into the other buffer while we compute on `cur`
    if (t + 1 < SS)
      stage_step(tid, row + 1, wsh[cur ^ 1], esh[cur ^ 1], ash[cur ^ 1],
                 wbuf, erase, addb);

    // read = w . mem  (threads 0..127, one output dim each)
    if (tid < DVDIM) {
      float r = 0.0f;
#pragma unroll 5
      for (int m = 0; m < MMEM; ++m) r = fmaf(wsh[cur][m], mem[m][tid], r);
      reads[row * DVDIM + tid] = r;
    }
    __syncthreads();  // read before update

    // mem = mem*(1 - f*w*e) + f*w*a   (== f*new + (1-f)*old)
    for (int i = tid; i < MMEM * DVDIM; i += 256) {
      int m = i / DVDIM, d = i % DVDIM;
      float wm = wsh[cur][m] * flag;
      mem[m][d] = fmaf(mem[m][d], 1.0f - wm * esh[cur][d], wm * ash[cur][d]);
    }
    // loop-top barrier protects update vs next stage re-targeting this buffer
  }
}

// ---------------- K4: h = tanh([reads,q_emb]@W_read + b_read); logits; BCE
#define X4PAD (192 + 4)
#define HPAD  (FDIM + 4)
__global__ __launch_bounds__(256) void k4_head(
    const float* __restrict__ reads, const int* __restrict__ qd,
    const float* __restrict__ qtab, const float* __restrict__ Wread,
    const float* __restrict__ bread, const float* __restrict__ Wpred,
    const float* __restrict__ bpred, const float* __restrict__ target,
    float* __restrict__ out, float* __restrict__ acc) {
  __shared__ float X[16][X4PAD];
  __shared__ float H[16][HPAD];
  const int row0 = blockIdx.x * 16;
  const int tid  = threadIdx.x;

  {  // stage concat([reads(128), q_emb(64)]) for 16 rows
    int rl = tid >> 4;
    int c0 = (tid & 15) * 8;
    const float* src = reads + (size_t)(row0 + rl) * DVDIM + c0;
#pragma unroll
    for (int j = 0; j < 8; ++j) X[rl][c0 + j] = src[j];
    int c1 = (tid & 15) * 4;
    const float* qs = qtab + (size_t)qd[row0 + rl] * DQDIM + c1;
#pragma unroll
    for (int j = 0; j < 4; ++j) X[rl][DVDIM + c1 + j] = qs[j];
  }
  __syncthreads();

  const int wave = tid >> 5, lane = tid & 31;
  const int n0 = wave * 16, mrow = lane & 15, khalf = (lane >> 4) * 2;
  v8f c = {};
  for (int ks = 0; ks < 192; ks += 4) {
    v2f a, b;
    a.x = X[mrow][ks + khalf];
    a.y = X[mrow][ks + khalf + 1];
    b.x = Wread[(size_t)(ks + khalf) * FDIM + n0 + mrow];
    b.y = Wread[(size_t)(ks + khalf + 1) * FDIM + n0 + mrow];
    c = wmma_k4(a, b, c);
  }
  {
    const int mbase = (lane >> 4) * 8;
    const int n = n0 + mrow;
#pragma unroll
    for (int r = 0; r < 8; ++r) H[mbase + r][n] = tanhf(c[r] + bread[n]);
  }
  __syncthreads();

  if (tid < 16) {
    const int row = row0 + tid;
    float lg = bpred[0];
    for (int n = 0; n < FDIM; ++n) lg = fmaf(H[tid][n], Wpred[n], lg);
    out[1 + row] = sigmoidf_(lg);
    float tgt = target[row];
    if (tgt >= 0.0f) {
      float bce = fmaxf(lg, 0.0f) - lg * tgt + log1pf(expf(-fabsf(lg)));
      atomicAdd(&acc[0], bce);
      atomicAdd(&acc[1], 1.0f);
    }
  }
}

// ---------------------------------------------------------------- K5: loss
__global__ void k5_loss(const float* __restrict__ acc, float* __restrict__ out) {
  if (threadIdx.x == 0) out[0] = acc[0] / fmaxf(acc[1], 1.0f);
}

extern "C" void kernel_launch(void* const* d_in, const int* in_sizes, int n_in,
                              void* d_out, int out_size, void* d_ws, size_t ws_size,
                              hipStream_t stream) {
  const int*   q_data    = (const int*)  d_in[0];
  const int*   qa_data   = (const int*)  d_in[1];
  const float* target    = (const float*)d_in[2];
  const float* q_table   = (const float*)d_in[3];
  const float* qa_table  = (const float*)d_in[4];
  const float* key_mem   = (const float*)d_in[5];
  const float* init_val  = (const float*)d_in[6];
  const float* W_e       = (const float*)d_in[7];
  const float* b_e       = (const float*)d_in[8];
  const float* W_a       = (const float*)d_in[9];
  const float* b_a       = (const float*)d_in[10];
  const float* W_read    = (const float*)d_in[11];
  const float* b_read    = (const float*)d_in[12];
  const float* W_pred    = (const float*)d_in[13];
  const float* b_pred    = (const float*)d_in[14];
  float* out = (float*)d_out;  // [0]=loss, [1..65536]=probs

  // workspace: w(64) + erase(128) + add(128) + reads(128) per row + 2 acc
  float* wbuf   = (float*)d_ws;                       // NROWS*MPAD
  float* erase  = wbuf  + (size_t)NROWS * MPAD;       // NROWS*DVDIM
  float* addb   = erase + (size_t)NROWS * DVDIM;      // NROWS*DVDIM
  float* reads  = addb  + (size_t)NROWS * DVDIM;      // NROWS*DVDIM
  float* acc    = reads + (size_t)NROWS * DVDIM;      // 2 floats
  // total: 65536*448*4 + 8 bytes ~= 117.4 MB

  k0_zero<<<1, 32, 0, stream>>>(acc);
  k1_softmax_w<<<NROWS / 256, 256, 0, stream>>>(q_data, q_table, key_mem, wbuf);
  k2_erase_add<<<NROWS / 16, 256, 0, stream>>>(qa_data, qa_table, W_e, b_e,
                                               W_a, b_a, erase, addb);
  k3_scan<<<BB, 256, 0, stream>>>(q_data, init_val, wbuf, erase, addb, reads);
  k4_head<<<NROWS / 16, 256, 0, stream>>>(reads, q_data, q_table, W_read, b_read,
                                          W_pred, b_pred, target, out, acc);
  k5_loss<<<1, 32, 0, stream>>>(acc, out);
}